// UnivariateHawkesProcess_67954972557800
// MI455X (gfx1250) — compile-verified
//
#include <hip/hip_runtime.h>
#include <hip/hip_bf16.h>
#include <stdint.h>

// ---------------------------------------------------------------------------
// Hawkes-process log-likelihood, MI455X (gfx1250, wave32).
//
// Roofline: 16 MB f32 stream @ 23.3 TB/s  => ~0.7 us memory floor;
//           ~2.1N transcendentals         => ~1 us VALU floor (dominant).
// Strategy: affine-recurrence scan R_i = d_i*R_{i-1} + 1 (d_i<=1, stable),
// tiled 4096 elems/block, async global->LDS staging (CDNA5 ASYNCcnt path),
// wave32 shuffle scans (barrier-light), backward-chained compensator exps,
// deterministic tree reductions (no atomics).
// ---------------------------------------------------------------------------

#define BLOCK  256
#define NWAVES (BLOCK / 32)
#define EPT    16
#define TILE   (BLOCK * EPT)         // 4096 floats = 16 KB per tile
#define TPT_B  8                     // tiles/thread in the tile-scan (<=2048 tiles)

static __device__ __forceinline__ void wait_async0() {
#if __has_builtin(__builtin_amdgcn_s_wait_asynccnt)
  __builtin_amdgcn_s_wait_asynccnt(0);
#else
  asm volatile("s_wait_asynccnt 0" ::: "memory");
#endif
}

// Stage one tile (TILE floats) from global into LDS via the CDNA5 async path.
// Full 16B chunks use global_load_async_to_lds_b128; the ragged tail (last
// tile only) falls back to guarded scalar LDS stores to avoid OOB reads.
static __device__ __forceinline__ void load_tile_async(const float* __restrict__ ts,
                                                       int N, int tileStart,
                                                       float* s_ts, int tid) {
  const uint32_t lds_base = (uint32_t)(uintptr_t)s_ts;   // low 32 bits = LDS byte offset
  const int NCHUNK = TILE / 4;                           // 16B chunks per tile
#pragma unroll
  for (int i = 0; i < NCHUNK / BLOCK; ++i) {
    const int c  = tid + i * BLOCK;
    const int g0 = tileStart + c * 4;
    if (g0 + 4 <= N) {
      const float*  gp = ts + g0;
      const uint32_t lp = lds_base + (uint32_t)(c * 16);
      asm volatile("global_load_async_to_lds_b128 %0, %1, off"
                   :: "v"(lp), "v"(gp) : "memory");
    } else {
#pragma unroll
      for (int e = 0; e < 4; ++e) {
        const int g = g0 + e;
        s_ts[c * 4 + e] = (g < N) ? ts[g] : 0.0f;
      }
    }
  }
  wait_async0();        // each wave drains its own ASYNCcnt
  __syncthreads();      // then cross-wave visibility
}

// Block-wide scan of affine maps R -> a*R + b (threads ordered by tid).
// wave32 shuffle scan (5 steps, no barriers) + 8 wave aggregates in LDS.
// Outputs: (Aexcl,Bexcl) = composition of threads 0..tid-1,
//          (Atot,Btot)   = composition of all BLOCK threads.
static __device__ __forceinline__ void affine_scan_block(float* sWA, float* sWB,
                                                         int tid, float A, float B,
                                                         float& Aexcl, float& Bexcl,
                                                         float& Atot,  float& Btot) {
  const int lane = tid & 31;
  const int wid  = tid >> 5;
  float a = A, b = B;
#pragma unroll
  for (int off = 1; off < 32; off <<= 1) {
    const float ap = __shfl_up(a, off, 32);
    const float bp = __shfl_up(b, off, 32);
    if (lane >= off) { b = a * bp + b; a = a * ap; }   // cur ∘ prev
  }
  if (lane == 31) { sWA[wid] = a; sWB[wid] = b; }      // wave aggregate
  __syncthreads();
  // wave-exclusive + block-total: serial ordered fold over 8 aggregates
  float Aw = 1.0f, Bw = 0.0f, At = 1.0f, Bt = 0.0f;
#pragma unroll
  for (int w = 0; w < NWAVES; ++w) {
    const float aw = sWA[w], bw = sWB[w];
    if (w < wid) { Bw = aw * Bw + bw; Aw = aw * Aw; }
    Bt = aw * Bt + bw; At = aw * At;
  }
  // lane-exclusive from wave-inclusive
  float al = __shfl_up(a, 1, 32);
  float bl = __shfl_up(b, 1, 32);
  if (lane == 0) { al = 1.0f; bl = 0.0f; }
  Aexcl = al * Aw;            // laneExcl ∘ waveExcl
  Bexcl = al * Bw + bl;
  Atot  = At;
  Btot  = Bt;
}

// ---- Kernel 1: per-tile meta (tLast[k], L[k] = tile-run of R from 0) -------
__global__ void __launch_bounds__(BLOCK)
hawkes_tile_meta(const float* __restrict__ ts, const float* __restrict__ p_lb,
                 int N, float* __restrict__ tLast, float* __restrict__ Ltile) {
  __shared__ __align__(16) float s_ts[TILE];
  __shared__ float sWA[NWAVES], sWB[NWAVES];
  const int k = blockIdx.x, tid = threadIdx.x;
  const float beta = __expf(p_lb[0]);
  const int tileStart = k * TILE;

  load_tile_async(ts, N, tileStart, s_ts, tid);

  const int base = tid * EPT;
  float tp = (tid == 0) ? s_ts[0] : s_ts[base - 1];  // thread0's A never used
  float A = 1.0f, B = 0.0f;
#pragma unroll
  for (int e = 0; e < EPT; ++e) {
    const int g = tileStart + base + e;
    const float t = s_ts[base + e];
    const float d = (g < N) ? __expf(-beta * (t - tp)) : 1.0f;
    A *= d;
    B = d * B + ((g < N) ? 1.0f : 0.0f);
    tp = t;
  }
  float Ae, Be, At, Bt;
  affine_scan_block(sWA, sWB, tid, A, B, Ae, Be, At, Bt);
  if (tid == 0) {
    Ltile[k] = Bt;                                  // tile run with R_in = 0
    const int lastIdx = min(tileStart + TILE, N) - 1;
    tLast[k] = s_ts[lastIdx - tileStart];
  }
}

// ---- Kernel 2: parallel affine scan across tiles -> carry Rprev[k] ---------
__global__ void __launch_bounds__(BLOCK)
hawkes_tile_scan(const float* __restrict__ p_lb, int nTiles,
                 const float* __restrict__ tLast, const float* __restrict__ Ltile,
                 float* __restrict__ Rprev) {
  __shared__ float sWA[NWAVES], sWB[NWAVES];
  const int tid = threadIdx.x;
  const float beta = __expf(p_lb[0]);
  const int k0 = tid * TPT_B;

  float aC[TPT_B], bC[TPT_B];
  float A = 1.0f, B = 0.0f;
  float prevT = 0.0f;
  if (k0 < nTiles) prevT = (k0 == 0) ? tLast[0] : tLast[k0 - 1];
#pragma unroll
  for (int i = 0; i < TPT_B; ++i) {
    const int k = k0 + i;
    float a = 1.0f, b = 0.0f;
    if (k < nTiles) {
      const float tl = tLast[k];
      a = __expf(-beta * (tl - prevT));   // k==0: exp(0)=1, multiplies R=0 anyway
      b = Ltile[k];
      prevT = tl;
    }
    aC[i] = a; bC[i] = b;
    A = A * a;
    B = a * B + b;
  }
  float Ae, Be, At, Bt;
  affine_scan_block(sWA, sWB, tid, A, B, Ae, Be, At, Bt);
  // exclusive carry applied to global R_in = 0  =>  R = Bexcl
  float R = Be;
#pragma unroll
  for (int i = 0; i < TPT_B; ++i) {
    const int k = k0 + i;
    if (k < nTiles) {
      Rprev[k] = R;
      R = aC[i] * R + bC[i];
    }
  }
}

// ---- Kernel 3: main pass — per-element intensity log + compensator ---------
__global__ void __launch_bounds__(BLOCK)
hawkes_main(const float* __restrict__ ts, const float* __restrict__ p_mu,
            const float* __restrict__ p_la, const float* __restrict__ p_lb,
            const int* __restrict__ p_T, int N,
            const float* __restrict__ Rprev,
            float* __restrict__ blockLL, float* __restrict__ blockCC) {
  __shared__ __align__(16) float s_ts[TILE];
  __shared__ float sWA[NWAVES], sWB[NWAVES];
  const int k = blockIdx.x, tid = threadIdx.x;
  const int lane = tid & 31, wid = tid >> 5;
  const float mu    = p_mu[0];
  const float alpha = __expf(p_la[0]);
  const float beta  = __expf(p_lb[0]);
  const float Tf    = (float)p_T[0];
  const float ab    = alpha * beta;
  const int tileStart = k * TILE;

  load_tile_async(ts, N, tileStart, s_ts, tid);

  const int base = tid * EPT;
  float tp;
  if (tid == 0) tp = (tileStart > 0) ? ts[tileStart - 1] : s_ts[0];
  else          tp = s_ts[base - 1];

  float d_e[EPT];
  float A = 1.0f, B = 0.0f;
#pragma unroll
  for (int e = 0; e < EPT; ++e) {
    const int g = tileStart + base + e;
    const float t = s_ts[base + e];
    const float d = (g < N) ? __expf(-beta * (t - tp)) : 1.0f;
    d_e[e] = d;
    A *= d;
    B = d * B + ((g < N) ? 1.0f : 0.0f);
    tp = t;
  }
  float Ae, Be, At, Bt;
  affine_scan_block(sWA, sWB, tid, A, B, Ae, Be, At, Bt);

  const float Rblk = Rprev[k];
  float R = Ae * Rblk + Be;

  // Forward walk: log-intensities via the decay chain.
  float accL = 0.0f;
#pragma unroll
  for (int e = 0; e < EPT; ++e) {
    const int g = tileStart + base + e;
    if (g < N) {
      const float S = d_e[e] * R;            // sum_{j<g} exp(-beta (t_g - t_j))
      R = S + 1.0f;
      accL += __logf(fmaf(ab, S, mu));       // log lambda(t_g)
    }
  }

  // Backward walk: compensator exp(-beta (T - t_i)) chained through d_e[],
  // one real exp per chunk (in-range elements are a prefix of the chunk).
  float accC = 0.0f;
  {
    float eT = 0.0f;
    bool init = false;
#pragma unroll
    for (int e = EPT - 1; e >= 0; --e) {
      const int g = tileStart + base + e;
      if (g < N) {
        if (!init) { eT = __expf(-beta * (Tf - s_ts[base + e])); init = true; }
        accC += 1.0f - eT;
        if (e > 0) eT *= d_e[e];             // exp(-b(T-t_{e-1})) = exp(-b(T-t_e))*d_e
      }
    }
  }

  // Deterministic block reduction: wave shuffle-reduce, then 8-entry LDS fold.
#pragma unroll
  for (int off = 16; off > 0; off >>= 1) {
    accL += __shfl_xor(accL, off, 32);
    accC += __shfl_xor(accC, off, 32);
  }
  __syncthreads();                           // sWA/sWB reads above are done
  if (lane == 0) { sWA[wid] = accL; sWB[wid] = accC; }
  __syncthreads();
  if (tid == 0) {
    float sL = 0.0f, sC = 0.0f;
#pragma unroll
    for (int w = 0; w < NWAVES; ++w) { sL += sWA[w]; sC += sWB[w]; }
    blockLL[k] = sL;
    blockCC[k] = sC;
  }
}

// ---- Kernel 4: deterministic final reduction -------------------------------
__global__ void __launch_bounds__(BLOCK)
hawkes_finalize(const float* __restrict__ p_mu, const float* __restrict__ p_la,
                const float* __restrict__ p_lb, const int* __restrict__ p_T,
                int nTiles, const float* __restrict__ blockLL,
                const float* __restrict__ blockCC, float* __restrict__ out) {
  __shared__ float sL[NWAVES], sC[NWAVES];
  const int tid = threadIdx.x;
  const int lane = tid & 31, wid = tid >> 5;
  float aL = 0.0f, aC = 0.0f;
  for (int i = tid; i < nTiles; i += BLOCK) { aL += blockLL[i]; aC += blockCC[i]; }
#pragma unroll
  for (int off = 16; off > 0; off >>= 1) {
    aL += __shfl_xor(aL, off, 32);
    aC += __shfl_xor(aC, off, 32);
  }
  if (lane == 0) { sL[wid] = aL; sC[wid] = aC; }
  __syncthreads();
  if (tid == 0) {
    float tL = 0.0f, tC = 0.0f;
#pragma unroll
    for (int w = 0; w < NWAVES; ++w) { tL += sL[w]; tC += sC[w]; }
    const float mu    = p_mu[0];
    const float alpha = __expf(p_la[0]);
    const float beta  = __expf(p_lb[0]);
    const float Tf    = (float)p_T[0];
    out[0] = tL - (mu * Tf + (alpha / beta) * tC);
  }
}

extern "C" void kernel_launch(void* const* d_in, const int* in_sizes, int n_in,
                              void* d_out, int out_size, void* d_ws, size_t ws_size,
                              hipStream_t stream) {
  const float* ts   = (const float*)d_in[0];
  const float* p_mu = (const float*)d_in[1];
  const float* p_la = (const float*)d_in[2];
  const float* p_lb = (const float*)d_in[3];
  const int*   p_T  = (const int*)d_in[4];
  const int N = in_sizes[0];
  const int nTiles = (N + TILE - 1) / TILE;   // 977 for N = 4M (<= 2048 supported)

  float* ws      = (float*)d_ws;
  float* tLast   = ws;
  float* Ltile   = ws + nTiles;
  float* Rprev   = ws + 2 * nTiles;
  float* blockLL = ws + 3 * nTiles;
  float* blockCC = ws + 4 * nTiles;

  hawkes_tile_meta<<<nTiles, BLOCK, 0, stream>>>(ts, p_lb, N, tLast, Ltile);
  hawkes_tile_scan<<<1, BLOCK, 0, stream>>>(p_lb, nTiles, tLast, Ltile, Rprev);
  hawkes_main<<<nTiles, BLOCK, 0, stream>>>(ts, p_mu, p_la, p_lb, p_T, N,
                                            Rprev, blockLL, blockCC);
  hawkes_finalize<<<1, BLOCK, 0, stream>>>(p_mu, p_la, p_lb, p_T, nTiles,
                                           blockLL, blockCC, (float*)d_out);
}